// WindowAttention3D_5377299054923
// MI455X (gfx1250) — compile-verified
//
#include <hip/hip_runtime.h>
#include <hip/hip_bf16.h>

typedef __attribute__((ext_vector_type(2))) float v2f;
typedef __attribute__((ext_vector_type(8))) float v8f;

#define NTOK 98
#define NPAD 112
#define CDIM 128
#define QKVC 384
#define HEADS 4
#define HD 32
#define NWIN 64

// LDS partition (in floats)
#define QKV_SZ  (NPAD*QKVC)          // 43008 (168 KB)
#define XO_SZ   (NPAD*CDIM)          // 14336 (56 KB)  x, later out-accum
#define ATT_SZ  (NPAD*NPAD)          // 12544 (49 KB)
#define LDS_FLOATS (QKV_SZ + XO_SZ + ATT_SZ)   // 69888 floats = 279,552 B < 320 KB

__global__ __launch_bounds__(256, 1)
void win_attn3d_kernel(const float* __restrict__ x,
                       const float* __restrict__ mask,
                       const float* __restrict__ qkv_w,
                       const float* __restrict__ qkv_b,
                       const float* __restrict__ proj_w,
                       const float* __restrict__ proj_b,
                       const float* __restrict__ bias_table,
                       const int*   __restrict__ rel_index,
                       float* __restrict__ out)
{
    extern __shared__ float smem[];
    float* s_qkv = smem;                    // [112][384]
    float* s_xo  = smem + QKV_SZ;           // [112][128]  x then out
    float* s_att = smem + QKV_SZ + XO_SZ;   // [112][112]

    const int b     = blockIdx.x;
    const int t     = threadIdx.x;
    const int wave  = t >> 5;
    const int lane  = t & 31;
    const int lan16 = lane & 15;
    const int half  = lane >> 4;            // which 16-lane half

    // ---------------- Phase 1: x -> LDS (zero-pad rows 98..111) ----------------
    const float* xb = x + (size_t)b * (NTOK * CDIM);
    for (int idx = t; idx < NPAD * CDIM; idx += 256)
        s_xo[idx] = (idx < NTOK * CDIM) ? xb[idx] : 0.0f;
    __syncthreads();

    // ---------------- Phase 2: qkv = x @ qkv_w^T + qkv_b  -> s_qkv -------------
    // 24 N-tiles of 16; wave w owns 3 of them, keeps B-frags (K=128) in registers
    for (int nti = 0; nti < 3; ++nti) {
        const int nt = wave * 3 + nti;
        const int n  = nt * 16 + lan16;
        v2f bf[32];
        #pragma unroll
        for (int ks = 0; ks < 32; ++ks) {
            const int kb = ks * 4 + half * 2;              // B[k][n] = qkv_w[n*128+k]
            bf[ks] = *(const v2f*)(qkv_w + n * CDIM + kb);
        }
        const float cinit = qkv_b[n];
        for (int mt = 0; mt < 7; ++mt) {
            v8f c;
            #pragma unroll
            for (int r = 0; r < 8; ++r) c[r] = cinit;
            const int m = mt * 16 + lan16;
            #pragma unroll
            for (int ks = 0; ks < 32; ++ks) {
                const int kb = ks * 4 + half * 2;
                v2f a = *(const v2f*)(s_xo + m * CDIM + kb);
                c = __builtin_amdgcn_wmma_f32_16x16x4_f32(false, a, false, bf[ks],
                                                          (short)0, c, false, false);
            }
            #pragma unroll
            for (int r = 0; r < 8; ++r)
                s_qkv[(mt * 16 + r + 8 * half) * QKVC + n] = c[r];
        }
    }
    __syncthreads();

    const float scale = 0.17677669529663687f;   // 32^-0.5
    const float* mwin = mask + (size_t)(b & (NWIN - 1)) * (NTOK * NTOK);

    // ---------------- Phase 3: per-head attention ------------------------------
    for (int h = 0; h < HEADS; ++h) {
        // attn = q @ k^T  (49 16x16 tiles, K = 32)
        for (int tile = wave; tile < 49; tile += 8) {
            const int it = tile / 7, jt = tile % 7;
            v8f c;
            #pragma unroll
            for (int r = 0; r < 8; ++r) c[r] = 0.0f;
            const int qrow = it * 16 + lan16;
            const int krow = jt * 16 + lan16;
            #pragma unroll
            for (int ks = 0; ks < 8; ++ks) {
                const int kb = ks * 4 + half * 2;
                v2f a  = *(const v2f*)(s_qkv + qrow * QKVC + h * HD + kb);
                v2f bb = *(const v2f*)(s_qkv + krow * QKVC + CDIM + h * HD + kb);
                c = __builtin_amdgcn_wmma_f32_16x16x4_f32(false, a, false, bb,
                                                          (short)0, c, false, false);
            }
            #pragma unroll
            for (int r = 0; r < 8; ++r)
                s_att[(it * 16 + r + 8 * half) * NPAD + jt * 16 + lan16] = c[r];
        }
        __syncthreads();

        // softmax rows, fusing scale + rel-pos-bias gather + mask (L2-resident)
        if (t < NTOK) {
            float* arow = s_att + t * NPAD;
            const int*   ri = rel_index + t * NTOK;
            const float* mr = mwin + t * NTOK;
            float mx = -3.0e38f;
            for (int j = 0; j < NTOK; ++j) {
                float s = arow[j] * scale + bias_table[ri[j] * HEADS + h] + mr[j];
                arow[j] = s;
                mx = fmaxf(mx, s);
            }
            float sum = 0.0f;
            for (int j = 0; j < NTOK; ++j) {
                float e = __expf(arow[j] - mx);
                arow[j] = e;
                sum += e;
            }
            const float inv = 1.0f / sum;
            for (int j = 0; j < NTOK; ++j) arow[j] *= inv;
            for (int j = NTOK; j < NPAD; ++j) arow[j] = 0.0f;   // pad cols -> exact 0
        } else if (t < NPAD) {
            float* arow = s_att + t * NPAD;
            for (int j = 0; j < NPAD; ++j) arow[j] = 0.0f;      // pad rows -> 0
        }
        __syncthreads();

        // out_h = attn @ v  (14 tiles: 7 M x 2 N, K = 112) -> s_xo cols [h*32, h*32+32)
        for (int tile = wave; tile < 14; tile += 8) {
            const int it = tile >> 1, nt2 = tile & 1;
            v8f c;
            #pragma unroll
            for (int r = 0; r < 8; ++r) c[r] = 0.0f;
            const int arow = it * 16 + lan16;
            const int vcol = 2 * CDIM + h * HD + nt2 * 16 + lan16;
            #pragma unroll
            for (int ks = 0; ks < 28; ++ks) {
                const int kb = ks * 4 + half * 2;
                v2f a = *(const v2f*)(s_att + arow * NPAD + kb);
                v2f bb;
                bb.x = s_qkv[kb * QKVC + vcol];
                bb.y = s_qkv[(kb + 1) * QKVC + vcol];
                c = __builtin_amdgcn_wmma_f32_16x16x4_f32(false, a, false, bb,
                                                          (short)0, c, false, false);
            }
            #pragma unroll
            for (int r = 0; r < 8; ++r)
                s_xo[(it * 16 + r + 8 * half) * CDIM + h * HD + nt2 * 16 + lan16] = c[r];
        }
        __syncthreads();
    }

    // ---------------- Phase 4: out = s_xo @ proj_w^T + proj_b -> global --------
    {
        const int n = wave * 16 + lan16;        // 8 waves x 16 = 128 output cols
        v2f bf[32];
        #pragma unroll
        for (int ks = 0; ks < 32; ++ks) {
            const int kb = ks * 4 + half * 2;
            bf[ks] = *(const v2f*)(proj_w + n * CDIM + kb);
        }
        const float cinit = proj_b[n];
        float* ob = out + (size_t)b * (NTOK * CDIM);
        for (int mt = 0; mt < 7; ++mt) {
            v8f c;
            #pragma unroll
            for (int r = 0; r < 8; ++r) c[r] = cinit;
            const int m = mt * 16 + lan16;
            #pragma unroll
            for (int ks = 0; ks < 32; ++ks) {
                const int kb = ks * 4 + half * 2;
                v2f a = *(const v2f*)(s_xo + m * CDIM + kb);
                c = __builtin_amdgcn_wmma_f32_16x16x4_f32(false, a, false, bf[ks],
                                                          (short)0, c, false, false);
            }
            #pragma unroll
            for (int r = 0; r < 8; ++r) {
                const int mr = mt * 16 + r + 8 * half;
                if (mr < NTOK) ob[mr * CDIM + n] = c[r];
            }
        }
    }
}

extern "C" void kernel_launch(void* const* d_in, const int* in_sizes, int n_in,
                              void* d_out, int out_size, void* d_ws, size_t ws_size,
                              hipStream_t stream) {
    (void)n_in; (void)out_size; (void)d_ws; (void)ws_size;
    const float* x    = (const float*)d_in[0];
    const float* mask = (const float*)d_in[1];
    const float* qw   = (const float*)d_in[2];
    const float* qb   = (const float*)d_in[3];
    const float* pw   = (const float*)d_in[4];
    const float* pb   = (const float*)d_in[5];
    const float* bt   = (const float*)d_in[6];
    const int*   ri   = (const int*)d_in[7];
    float* out = (float*)d_out;

    const int nB = in_sizes[0] / (NTOK * CDIM);          // 2048 windows
    const size_t lds_bytes = (size_t)LDS_FLOATS * sizeof(float);

    // Allow >64KB dynamic LDS (WGP supports up to 320KB per workgroup on CDNA5).
    static int attr_set = 0;
    if (!attr_set) {
        hipFuncSetAttribute((const void*)win_attn3d_kernel,
                            hipFuncAttributeMaxDynamicSharedMemorySize,
                            (int)lds_bytes);
        attr_set = 1;
    }

    win_attn3d_kernel<<<nB, 256, lds_bytes, stream>>>(x, mask, qw, qb, pw, pb, bt, ri, out);
}